// GCNConv_83253646065701
// MI455X (gfx1250) — compile-verified
//
#include <hip/hip_runtime.h>
#include <hip/hip_bf16.h>

typedef __attribute__((ext_vector_type(2))) float v2f;
typedef __attribute__((ext_vector_type(8))) float v8f;

#define D_FEAT 256

// ---------------------------------------------------------------------------
// 1) degree: deg[row[e]] += 1  (float atomics; counts are small -> exact)
// ---------------------------------------------------------------------------
__global__ __launch_bounds__(256) void deg_kernel(const int* __restrict__ row,
                                                  float* __restrict__ deg,
                                                  int n_edges) {
    int e = blockIdx.x * blockDim.x + threadIdx.x;
    if (e < n_edges) {
        atomicAdd(&deg[row[e]], 1.0f);
    }
}

// ---------------------------------------------------------------------------
// 2) dinv[i] = deg>0 ? deg^-0.5 : 0   (in place)
// ---------------------------------------------------------------------------
__global__ __launch_bounds__(256) void dinv_kernel(float* __restrict__ deg,
                                                   int n_nodes) {
    int i = blockIdx.x * blockDim.x + threadIdx.x;
    if (i < n_nodes) {
        float d = deg[i];
        deg[i] = (d > 0.0f) ? rsqrtf(d) : 0.0f;
    }
}

// ---------------------------------------------------------------------------
// 3) GEMM: h[n, o] = sum_k x[n,k] * W[o,k] + b[o]
//    Full-precision f32 WMMA (V_WMMA_F32_16X16X4_F32).
//    A-frag layout: lane(l<16) holds (M=l, K=k0+0/1); lane(l+16) holds (M=l, K=k0+2/3)
//      -> float2 load from row-major x.
//    B-frag (mirror): lane holds (N=lane&15, same K pair)
//      -> float2 load from W[out][in] row-major. No transposes needed.
//    Wave computes a 16x64 strip (4 accumulators); 8 waves/block = 32 rows x 256 cols.
//    100000 rows = 6250 exact 16-row tiles -> EXEC all-ones everywhere (WMMA req).
// ---------------------------------------------------------------------------
__global__ __launch_bounds__(256) void gemm_wmma_kernel(const float* __restrict__ x,
                                                        const float* __restrict__ W,
                                                        const float* __restrict__ bias,
                                                        float* __restrict__ h,
                                                        int n_mtiles) {
    const int lane  = threadIdx.x & 31;
    const int wave  = threadIdx.x >> 5;                 // 0..7
    const int mtile = blockIdx.x * 2 + (wave >> 2);     // 2 m-tiles per block
    if (mtile >= n_mtiles) return;                      // wave-uniform guard
    const int n0    = (wave & 3) * 64;                  // n strip base: 0/64/128/192
    const int m0    = mtile * 16;
    const int l15   = lane & 15;
    const int khalf = lane >> 4;                        // 0 or 1 -> K pair select

    const float* xrow = x + (size_t)(m0 + l15) * D_FEAT;
    const float* wp0  = W + (size_t)(n0 +  0 + l15) * D_FEAT;
    const float* wp1  = W + (size_t)(n0 + 16 + l15) * D_FEAT;
    const float* wp2  = W + (size_t)(n0 + 32 + l15) * D_FEAT;
    const float* wp3  = W + (size_t)(n0 + 48 + l15) * D_FEAT;

    v8f acc0 = {}, acc1 = {}, acc2 = {}, acc3 = {};

    #pragma unroll 4
    for (int k = 0; k < D_FEAT; k += 4) {
        const int kb = k + khalf * 2;
        v2f a  = *(const v2f*)(xrow + kb);
        v2f b0 = *(const v2f*)(wp0 + kb);
        v2f b1 = *(const v2f*)(wp1 + kb);
        v2f b2 = *(const v2f*)(wp2 + kb);
        v2f b3 = *(const v2f*)(wp3 + kb);
        acc0 = __builtin_amdgcn_wmma_f32_16x16x4_f32(false, a, false, b0, (short)0, acc0, false, false);
        acc1 = __builtin_amdgcn_wmma_f32_16x16x4_f32(false, a, false, b1, (short)0, acc1, false, false);
        acc2 = __builtin_amdgcn_wmma_f32_16x16x4_f32(false, a, false, b2, (short)0, acc2, false, false);
        acc3 = __builtin_amdgcn_wmma_f32_16x16x4_f32(false, a, false, b3, (short)0, acc3, false, false);
    }

    // D layout: VGPR r -> (M = r + 8*khalf, N = lane&15) within the 16x16 tile.
    const int mbase = m0 + khalf * 8;
    v8f accs[4] = {acc0, acc1, acc2, acc3};
    #pragma unroll
    for (int j = 0; j < 4; ++j) {
        const int nj = n0 + j * 16 + l15;
        const float bj = bias[nj];
        float* hp = h + (size_t)mbase * D_FEAT + nj;
        #pragma unroll
        for (int r = 0; r < 8; ++r) {
            hp[(size_t)r * D_FEAT] = accs[j][r] + bj;
        }
    }
}

// ---------------------------------------------------------------------------
// 4) SpMM scatter: out[row[e]] += dinv[col[e]] * h[col[e]]
//    One wave per edge; lane covers 8 contiguous features (2x float4 gathers,
//    8 f32 atomics). h (102 MB) is L2-resident on MI455X (192 MB L2), so the
//    gather + atomic traffic stays on-chip.
// ---------------------------------------------------------------------------
__global__ __launch_bounds__(256) void spmm_kernel(const int* __restrict__ row,
                                                   const int* __restrict__ col,
                                                   const float* __restrict__ dinv,
                                                   const float* __restrict__ h,
                                                   float* __restrict__ out,
                                                   int n_edges) {
    const int lane = threadIdx.x & 31;
    const int e    = blockIdx.x * 8 + (threadIdx.x >> 5);
    if (e >= n_edges) return;

    const int r = row[e];
    const int c = col[e];
    const float w = dinv[c];

    const float4* hp = (const float4*)(h + (size_t)c * D_FEAT + lane * 8);
    float4 v0 = hp[0];
    float4 v1 = hp[1];

    float* op = out + (size_t)r * D_FEAT + lane * 8;
    atomicAdd(op + 0, w * v0.x);
    atomicAdd(op + 1, w * v0.y);
    atomicAdd(op + 2, w * v0.z);
    atomicAdd(op + 3, w * v0.w);
    atomicAdd(op + 4, w * v1.x);
    atomicAdd(op + 5, w * v1.y);
    atomicAdd(op + 6, w * v1.z);
    atomicAdd(op + 7, w * v1.w);
}

// ---------------------------------------------------------------------------
// 5) final scale: out[i,:] *= dinv[i]   (applies the dinv[row] factor)
// ---------------------------------------------------------------------------
__global__ __launch_bounds__(256) void scale_kernel(float* __restrict__ out,
                                                    const float* __restrict__ dinv,
                                                    int n_nodes) {
    size_t idx = (size_t)blockIdx.x * blockDim.x + threadIdx.x;
    if (idx < (size_t)n_nodes * D_FEAT) {
        out[idx] *= dinv[idx >> 8];   // D_FEAT == 256
    }
}

// ---------------------------------------------------------------------------
// launcher
// ---------------------------------------------------------------------------
extern "C" void kernel_launch(void* const* d_in, const int* in_sizes, int n_in,
                              void* d_out, int out_size, void* d_ws, size_t ws_size,
                              hipStream_t stream) {
    const float* x    = (const float*)d_in[0];
    const int*   ei   = (const int*)d_in[1];
    const float* W    = (const float*)d_in[2];
    const float* bias = (const float*)d_in[3];
    float*       out  = (float*)d_out;

    const int n_nodes = in_sizes[0] / D_FEAT;   // 100000
    const int n_edges = in_sizes[1] / 2;        // 3200000
    const int* row = ei;
    const int* col = ei + n_edges;

    // workspace: [dinv: n_nodes floats][h: n_nodes*256 floats]
    char* ws = (char*)d_ws;
    float* deg = (float*)ws;  // becomes dinv in place
    size_t h_off = (((size_t)n_nodes * sizeof(float) + 1023) / 1024) * 1024;
    float* h = (float*)(ws + h_off);

    // zero accumulators every call (d_out is poisoned; ws state must not persist)
    hipMemsetAsync(deg, 0, (size_t)n_nodes * sizeof(float), stream);
    hipMemsetAsync(out, 0, (size_t)out_size * sizeof(float), stream);

    deg_kernel<<<(n_edges + 255) / 256, 256, 0, stream>>>(row, deg, n_edges);
    dinv_kernel<<<(n_nodes + 255) / 256, 256, 0, stream>>>(deg, n_nodes);

    const int n_mtiles = n_nodes / 16;          // 6250 (exact)
    gemm_wmma_kernel<<<(n_mtiles + 1) / 2, 256, 0, stream>>>(x, W, bias, h, n_mtiles);

    spmm_kernel<<<(n_edges + 7) / 8, 256, 0, stream>>>(row, col, deg, h, out, n_edges);
    scale_kernel<<<((size_t)n_nodes * D_FEAT + 255) / 256, 256, 0, stream>>>(out, deg, n_nodes);
}